// PCCN_21345987461168
// MI455X (gfx1250) — compile-verified
//
#include <hip/hip_runtime.h>

typedef __attribute__((ext_vector_type(2))) float v2f;
typedef __attribute__((ext_vector_type(8))) float v8f;

#define BQ 8
#define NQ 8192
#define KQ 16
#define IQ 64
#define OQ 64
#define TILE_N 8
#define THREADS 128
#define SSTR 33          // padded row stride of the per-wave transpose scratch

__global__ __launch_bounds__(THREADS)
__attribute__((amdgpu_waves_per_eu(4)))     // cap at 256 VGPRs: no MSB mode, 4 waves/SIMD
void pccn_wmma(const float* __restrict__ input,
               const float* __restrict__ points,
               const float* __restrict__ support,
               const float* __restrict__ weight,
               const float* __restrict__ bias,
               const float* __restrict__ W1, const float* __restrict__ b1,
               const float* __restrict__ W2, const float* __restrict__ b2,
               const float* __restrict__ W3, const float* __restrict__ b3,
               float* __restrict__ out, float* __restrict__ out_sp)
{
    __shared__ float sWt[IQ * OQ];          // 16 KB  projection weight (row-major)
    __shared__ float sW3[32 * 64];          //  8 KB  W3
    __shared__ float sW2[16 * 32];          //  2 KB  W2
    __shared__ float sScr[4][16 * SSTR];    //  8.4 KB per-wave D->A bounce

    const int tid  = threadIdx.x;
    const int blk  = blockIdx.x;
    const int b    = blk >> 10;             // 1024 blocks per batch
    const int n0   = (blk & 1023) * TILE_N;
    const int wave = tid >> 5;
    const int lane = tid & 31;
    const int j    = lane & 15;             // tile column (= o%16) and k index
    const int half = lane >> 4;             // 0/1
    const int kk2  = 2 * half;              // K offset of this half inside a K=4 chunk

    // tuple output #2: support_points passthrough
    {
        int gid = blk * THREADS + tid;
        if (gid < BQ * 3 * NQ) out_sp[gid] = support[gid];
    }

    // stage weights into LDS (vectorized b128 copies)
    {
        const float4* s0 = (const float4*)weight;
        float4* d0 = (float4*)sWt;
        for (int i = tid; i < IQ * OQ / 4; i += THREADS) d0[i] = s0[i];
        const float4* s1 = (const float4*)W3;
        float4* d1 = (float4*)sW3;
        for (int i = tid; i < 32 * 64 / 4; i += THREADS) d1[i] = s1[i];
        const float4* s2 = (const float4*)W2;
        float4* d2 = (float4*)sW2;
        for (int i = tid; i < 16 * 32 / 4; i += THREADS) d2[i] = s2[i];
    }
    __syncthreads();

    // ---------- small hoisted per-lane constants ----------
    v2f w1f;                                  // W1: (3,16), row 3 zero-padded
    w1f.x = half ? W1[2 * 16 + j] : W1[0 * 16 + j];
    w1f.y = half ? 0.0f           : W1[1 * 16 + j];
    const float b1v = b1[j];
    float b2v[2], b3v[4], bov[4];
    #pragma unroll
    for (int t = 0; t < 2; ++t) b2v[t] = b2[t * 16 + j];
    #pragma unroll
    for (int t = 0; t < 4; ++t) { b3v[t] = b3[t * 16 + j]; bov[t] = bias[t * 16 + j]; }

    float* scr = sScr[wave];

    #pragma unroll 1
    for (int it = 0; it < 2; ++it) {
        const int n = n0 + wave * 2 + it;

        // ---- geometry + normalization (all 32 lanes duplicate the 16 k's) ----
        const int sp = (b * 3) * NQ + n;
        const int pp = sp * KQ + j;
        float p0 = points[pp]               - support[sp];
        float p1 = points[pp +     NQ * KQ] - support[sp +     NQ];
        float p2 = points[pp + 2 * NQ * KQ] - support[sp + 2 * NQ];
        float sq = p0 * p0 + p1 * p1 + p2 * p2;
        #pragma unroll
        for (int m = 8; m >= 1; m >>= 1) sq = fmaxf(sq, __shfl_xor(sq, m));
        float maxi = sqrtf(sq);
        maxi = (maxi == 0.0f) ? 1.0f : maxi;
        const float inv = 1.0f / maxi;

        // X A-frag: rows m=k, cols = d (d=3 zero-padded to K=4)
        v2f xf;
        xf.x = half ? p2 * inv : p0 * inv;
        xf.y = half ? 0.0f     : p1 * inv;

        // ---- h1 = relu(X @ W1 + b1): one WMMA, bias in C ----
        v8f h1;
        #pragma unroll
        for (int v = 0; v < 8; ++v) h1[v] = b1v;
        h1 = __builtin_amdgcn_wmma_f32_16x16x4_f32(false, xf, false, w1f,
                                                   (short)0, h1, false, false);
        #pragma unroll
        for (int v = 0; v < 8; ++v) h1[v] = fmaxf(h1[v], 0.0f);

        // bounce h1 (D layout) -> A layout via per-wave LDS scratch
        #pragma unroll
        for (int v = 0; v < 8; ++v) scr[(v + 8 * half) * SSTR + j] = h1[v];
        v2f h1A[4];
        #pragma unroll
        for (int c = 0; c < 4; ++c) {
            int col = 4 * c + kk2;
            h1A[c].x = scr[j * SSTR + col];
            h1A[c].y = scr[j * SSTR + col + 1];
        }

        // ---- h2 = relu(h1 @ W2 + b2): 2 tiles, interleaved chains ----
        v8f h2t[2];
        #pragma unroll
        for (int t = 0; t < 2; ++t)
            #pragma unroll
            for (int v = 0; v < 8; ++v) h2t[t][v] = b2v[t];
        #pragma unroll
        for (int c = 0; c < 4; ++c) {
            int r = 4 * c + kk2;
            #pragma unroll
            for (int t = 0; t < 2; ++t) {
                v2f wf;
                wf.x = sW2[r * 32 + t * 16 + j];
                wf.y = sW2[(r + 1) * 32 + t * 16 + j];
                h2t[t] = __builtin_amdgcn_wmma_f32_16x16x4_f32(false, h1A[c], false, wf,
                                                               (short)0, h2t[t], false, false);
            }
        }
        #pragma unroll
        for (int t = 0; t < 2; ++t)
            #pragma unroll
            for (int v = 0; v < 8; ++v) h2t[t][v] = fmaxf(h2t[t][v], 0.0f);

        // bounce h2 -> A layout (cols 0..31)
        #pragma unroll
        for (int t = 0; t < 2; ++t)
            #pragma unroll
            for (int v = 0; v < 8; ++v)
                scr[(v + 8 * half) * SSTR + t * 16 + j] = h2t[t][v];
        v2f h2A[8];
        #pragma unroll
        for (int c = 0; c < 8; ++c) {
            int col = 4 * c + kk2;
            h2A[c].x = scr[j * SSTR + col];
            h2A[c].y = scr[j * SSTR + col + 1];
        }

        // ---- mat = h2 @ W3 + b3: 4 o-tiles, interleaved chains ----
        v8f mt[4];
        #pragma unroll
        for (int t = 0; t < 4; ++t)
            #pragma unroll
            for (int v = 0; v < 8; ++v) mt[t][v] = b3v[t];
        #pragma unroll
        for (int c = 0; c < 8; ++c) {
            int r = 4 * c + kk2;
            #pragma unroll
            for (int t = 0; t < 4; ++t) {
                v2f wf;
                wf.x = sW3[r * 64 + t * 16 + j];
                wf.y = sW3[(r + 1) * 64 + t * 16 + j];
                mt[t] = __builtin_amdgcn_wmma_f32_16x16x4_f32(false, h2A[c], false, wf,
                                                              (short)0, mt[t], false, false);
            }
        }

        // ---- feats = input @ weight: K=64 over 16 chunks, 4 o-tiles ----
        // input is a 256 MB single-use stream -> non-temporal loads
        const float* Ab = input + (size_t)b * (IQ * NQ * KQ)
                                + (size_t)kk2 * (NQ * KQ) + n * KQ + j;
        v8f ft[4];
        #pragma unroll
        for (int t = 0; t < 4; ++t)
            #pragma unroll
            for (int v = 0; v < 8; ++v) ft[t][v] = 0.0f;
        #pragma unroll 8
        for (int c = 0; c < 16; ++c) {
            v2f a;
            a.x = __builtin_nontemporal_load(Ab + (size_t)(4 * c)     * (NQ * KQ));
            a.y = __builtin_nontemporal_load(Ab + (size_t)(4 * c + 1) * (NQ * KQ));
            int r = 4 * c + kk2;
            #pragma unroll
            for (int t = 0; t < 4; ++t) {
                v2f wf;
                wf.x = sWt[r * 64 + t * 16 + j];
                wf.y = sWt[(r + 1) * 64 + t * 16 + j];
                ft[t] = __builtin_amdgcn_wmma_f32_16x16x4_f32(false, a, false, wf,
                                                              (short)0, ft[t], false, false);
            }
        }

        // ---- fused reduction over k: D-tiles of ft and mt are layout-aligned ----
        #pragma unroll
        for (int t = 0; t < 4; ++t) {
            float partial = 0.0f;
            #pragma unroll
            for (int v = 0; v < 8; ++v) partial = fmaf(ft[t][v], mt[t][v], partial);
            partial += __shfl_xor(partial, 16);
            if (half == 0)
                __builtin_nontemporal_store(partial + bov[t],
                    out + ((size_t)b * OQ + t * 16 + j) * NQ + n);
        }
    }
}

extern "C" void kernel_launch(void* const* d_in, const int* in_sizes, int n_in,
                              void* d_out, int out_size, void* d_ws, size_t ws_size,
                              hipStream_t stream) {
    const float* input   = (const float*)d_in[0];
    const float* points  = (const float*)d_in[1];
    const float* support = (const float*)d_in[2];
    const float* weight  = (const float*)d_in[3];
    const float* bias    = (const float*)d_in[4];
    const float* W1 = (const float*)d_in[5];
    const float* b1 = (const float*)d_in[6];
    const float* W2 = (const float*)d_in[7];
    const float* b2 = (const float*)d_in[8];
    const float* W3 = (const float*)d_in[9];
    const float* b3 = (const float*)d_in[10];

    float* out    = (float*)d_out;
    float* out_sp = out + (size_t)BQ * OQ * NQ;   // tuple output #2

    dim3 grid(BQ * (NQ / TILE_N));                // 8192 blocks
    pccn_wmma<<<grid, THREADS, 0, stream>>>(input, points, support, weight, bias,
                                            W1, b1, W2, b2, W3, b3, out, out_sp);
}